// Mesh_Reduced_73263552135493
// MI455X (gfx1250) — compile-verified
//
#include <hip/hip_runtime.h>
#include <cstdint>

#define N_MESH  20000
#define N_EDGES 100000
#define N_PIV   256
#define HD      128

typedef __attribute__((ext_vector_type(2))) float v2f;
typedef __attribute__((ext_vector_type(8))) float v8f;

__device__ __forceinline__ float4 ld4(const float* p) { return *(const float4*)p; }

// ---------------------------------------------------------------- utilities
__global__ __launch_bounds__(256) void fill_zero_kernel(float* __restrict__ p, long n) {
    long i = (long)blockIdx.x * blockDim.x + threadIdx.x;
    long stride = (long)gridDim.x * blockDim.x;
    for (; i < n; i += stride) p[i] = 0.f;
}

__global__ __launch_bounds__(256) void deg_kernel(const int* __restrict__ dst,
                                                  float* __restrict__ deg, int nE) {
    int e = blockIdx.x * 256 + threadIdx.x;
    if (e < nE)
        __hip_atomic_fetch_add(&deg[dst[e]], 1.0f, __ATOMIC_RELAXED, __HIP_MEMORY_SCOPE_AGENT);
}

__global__ __launch_bounds__(256) void invdeg_kernel(float* __restrict__ deg, int n) {
    int i = blockIdx.x * 256 + threadIdx.x;
    if (i < n) deg[i] = 1.0f / fmaxf(deg[i], 1.0f);
}

__global__ __launch_bounds__(256) void scatter_add_kernel(const float* __restrict__ ef,
                                                          const int* __restrict__ dst,
                                                          float* __restrict__ agg, long total) {
    long i = (long)blockIdx.x * 256 + threadIdx.x;
    if (i >= total) return;
    int e = (int)(i >> 7);
    int c = (int)(i & 127);
    __hip_atomic_fetch_add(&agg[(size_t)dst[e] * HD + c], ef[i],
                           __ATOMIC_RELAXED, __HIP_MEMORY_SCOPE_AGENT);
}

// ------------------------------------------------- small-K / small-N layers
// Y[M,128] = relu(X[M,3] @ W[3,128] + b)
__global__ __launch_bounds__(256) void gemm_k3_kernel(const float* __restrict__ X,
                                                      const float* __restrict__ W,
                                                      const float* __restrict__ b,
                                                      float* __restrict__ Y, int M) {
    int m = blockIdx.x * 2 + (threadIdx.x >> 7);
    int n = threadIdx.x & 127;
    if (m >= M) return;
    float x0 = X[(size_t)m * 3 + 0];
    float x1 = X[(size_t)m * 3 + 1];
    float x2 = X[(size_t)m * 3 + 2];
    float v = b[n] + x0 * W[n] + x1 * W[HD + n] + x2 * W[2 * HD + n];
    Y[(size_t)m * HD + n] = fmaxf(v, 0.f);
}

// Y[M,3] = X[M,128] @ W[128,3] + b   (no activation)
__global__ __launch_bounds__(256) void gemm_n3_kernel(const float* __restrict__ X,
                                                      const float* __restrict__ W,
                                                      const float* __restrict__ b,
                                                      float* __restrict__ Y, int M) {
    int m = blockIdx.x * 256 + threadIdx.x;
    if (m >= M) return;
    float a0 = b[0], a1 = b[1], a2 = b[2];
    const float* x = X + (size_t)m * HD;
#pragma unroll 8
    for (int k = 0; k < HD; ++k) {
        float xv = x[k];
        a0 += xv * W[k * 3 + 0];
        a1 += xv * W[k * 3 + 1];
        a2 += xv * W[k * 3 + 2];
    }
    Y[(size_t)m * 3 + 0] = a0;
    Y[(size_t)m * 3 + 1] = a1;
    Y[(size_t)m * 3 + 2] = a2;
}

// ---------------------------------------------------------------- WMMA GEMM
// Y[M,128] = act(X[M,K] @ W[K,128] + bias)
// GATHER==0 : X = X0 (dense, row stride K)
// GATHER==2 : X = [ X0[m] | X1[m] * invdeg[m] ]              (K=256)
// GATHER==3 : X = [ X0[m] | X1[src[m]] | X1[dst[m]] ]        (K=384)
template <int GATHER, int K, bool RELU>
__global__ __launch_bounds__(256) void wmma_mlp_kernel(
    const float* __restrict__ X0, const float* __restrict__ X1,
    const int* __restrict__ srcIdx, const int* __restrict__ dstIdx,
    const float* __restrict__ invdeg,
    const float* __restrict__ W, const float* __restrict__ bias,
    float* __restrict__ Y, int M) {
    constexpr int LDK = K + 4;  // +4 float pad: kills LDS bank conflicts on b64 reads
    __shared__ float Xs[64 * LDK];
    __shared__ int sS[64], sD[64];
    __shared__ float sInv[64];

    const int tid = threadIdx.x;
    const int m0  = blockIdx.x * 64;

    if (GATHER != 0) {
        if (tid < 64) {
            int m = m0 + tid;
            if (GATHER == 3) {
                sS[tid] = (m < M) ? srcIdx[m] : 0;
                sD[tid] = (m < M) ? dstIdx[m] : 0;
            } else {
                sInv[tid] = (m < M) ? invdeg[m] : 0.f;
            }
        }
        __syncthreads();
    }

    // ---- stage activation tile (with fused gather/concat) into LDS
    constexpr int C4 = K / 4;
    for (int i = tid; i < 64 * C4; i += 256) {
        int row = i / C4;
        int c4  = (i % C4) * 4;
        int m   = m0 + row;
        float4 v = make_float4(0.f, 0.f, 0.f, 0.f);
        if (m < M) {
            if (GATHER == 0) {
                v = ld4(X0 + (size_t)m * K + c4);
            } else if (GATHER == 2) {
                if (c4 < HD) {
                    v = ld4(X0 + (size_t)m * HD + c4);
                } else {
                    v = ld4(X1 + (size_t)m * HD + (c4 - HD));
                    float s = sInv[row];
                    v.x *= s; v.y *= s; v.z *= s; v.w *= s;
                }
            } else {
                if (c4 < HD)            v = ld4(X0 + (size_t)m * HD + c4);
                else if (c4 < 2 * HD)   v = ld4(X1 + (size_t)sS[row] * HD + (c4 - HD));
                else                    v = ld4(X1 + (size_t)sD[row] * HD + (c4 - 2 * HD));
            }
        }
        *(float4*)&Xs[row * LDK + c4] = v;
    }
    __syncthreads();

    const int lane = tid & 31;
    const int wave = tid >> 5;
    const int n0   = wave * 16;   // 8 waves x 16 cols = 128 output cols
    const int ln   = lane & 15;
    const int hi   = lane >> 4;   // 0: K+0/1   1: K+2/3  (ISA 16x4 f32 A layout)

    v8f acc0 = {}, acc1 = {}, acc2 = {}, acc3 = {};

    const float* Wcol  = W + n0 + ln;
    const float* xrow0 = &Xs[(ln +  0) * LDK];
    const float* xrow1 = &Xs[(ln + 16) * LDK];
    const float* xrow2 = &Xs[(ln + 32) * LDK];
    const float* xrow3 = &Xs[(ln + 48) * LDK];

#pragma unroll 4
    for (int k0 = 0; k0 < K; k0 += 4) {
        int kb = k0 + hi * 2;
        v2f bf;
        bf.x = Wcol[(size_t)kb * HD];
        bf.y = Wcol[(size_t)(kb + 1) * HD];
        v2f a0 = *(const v2f*)(xrow0 + kb);
        v2f a1 = *(const v2f*)(xrow1 + kb);
        v2f a2 = *(const v2f*)(xrow2 + kb);
        v2f a3 = *(const v2f*)(xrow3 + kb);
        acc0 = __builtin_amdgcn_wmma_f32_16x16x4_f32(false, a0, false, bf, (short)0, acc0, false, false);
        acc1 = __builtin_amdgcn_wmma_f32_16x16x4_f32(false, a1, false, bf, (short)0, acc1, false, false);
        acc2 = __builtin_amdgcn_wmma_f32_16x16x4_f32(false, a2, false, bf, (short)0, acc2, false, false);
        acc3 = __builtin_amdgcn_wmma_f32_16x16x4_f32(false, a3, false, bf, (short)0, acc3, false, false);
    }

    int colg = n0 + ln;
    float bv = bias[colg];
    auto store_tile = [&](v8f a, int ms) {
#pragma unroll
        for (int r = 0; r < 8; ++r) {
            int rowg = m0 + ms * 16 + hi * 8 + r;
            if (rowg < M) {
                float v = a[r] + bv;
                if (RELU) v = fmaxf(v, 0.f);
                Y[(size_t)rowg * HD + colg] = v;
            }
        }
    };
    store_tile(acc0, 0);
    store_tile(acc1, 1);
    store_tile(acc2, 2);
    store_tile(acc3, 3);
}

// ---------------------------------------------------------------- LayerNorm
// one wave (32 lanes) per 128-wide row; optional residual add
template <bool RES>
__global__ __launch_bounds__(256) void ln128_kernel(const float* __restrict__ X,
                                                    const float* __restrict__ g,
                                                    const float* __restrict__ be,
                                                    const float* __restrict__ res,
                                                    float* __restrict__ Y, int M) {
    int lane = threadIdx.x & 31;
    int m = blockIdx.x * 8 + (threadIdx.x >> 5);
    if (m >= M) return;
    const float* x = X + (size_t)m * HD;
    float4 v = ld4(x + lane * 4);
    float s  = v.x + v.y + v.z + v.w;
    float s2 = v.x * v.x + v.y * v.y + v.z * v.z + v.w * v.w;
#pragma unroll
    for (int o = 16; o > 0; o >>= 1) {
        s  += __shfl_xor(s, o, 32);
        s2 += __shfl_xor(s2, o, 32);
    }
    float mean = s * (1.f / HD);
    float var  = s2 * (1.f / HD) - mean * mean;
    float inv  = rsqrtf(var + 1e-5f);
    float vals[4] = {v.x, v.y, v.z, v.w};
#pragma unroll
    for (int j = 0; j < 4; ++j) {
        int c = lane * 4 + j;
        float o2 = (vals[j] - mean) * inv * g[c] + be[c];
        if (RES) o2 += res[(size_t)m * HD + c];
        Y[(size_t)m * HD + c] = o2;
    }
}

// LayerNorm over last dim = 3 (in place)
__global__ __launch_bounds__(256) void ln3_kernel(float* __restrict__ Z,
                                                  const float* __restrict__ g,
                                                  const float* __restrict__ be, int M) {
    int m = blockIdx.x * 256 + threadIdx.x;
    if (m >= M) return;
    float a = Z[(size_t)m * 3 + 0], b = Z[(size_t)m * 3 + 1], c = Z[(size_t)m * 3 + 2];
    float mean = (a + b + c) * (1.f / 3.f);
    float va = ((a - mean) * (a - mean) + (b - mean) * (b - mean) + (c - mean) * (c - mean)) * (1.f / 3.f);
    float inv = rsqrtf(va + 1e-5f);
    Z[(size_t)m * 3 + 0] = (a - mean) * inv * g[0] + be[0];
    Z[(size_t)m * 3 + 1] = (b - mean) * inv * g[1] + be[1];
    Z[(size_t)m * 3 + 2] = (c - mean) * inv * g[2] + be[2];
}

// ---------------------------------------------------------------- KNN interp
// pivotal <- mesh : one block per pivot, cooperative top-3 over 20000 points
__global__ __launch_bounds__(256) void knn_m2p_kernel(const float* __restrict__ z,
                                                      const float* __restrict__ pm,
                                                      const float* __restrict__ pp,
                                                      float* __restrict__ zpiv) {
    __shared__ float sd[256 * 3];
    __shared__ int   si[256 * 3];
    int p = blockIdx.x, tid = threadIdx.x;
    float py0 = pp[p * 2 + 0], py1 = pp[p * 2 + 1];
    float da = 3.4e38f, db = 3.4e38f, dc = 3.4e38f;
    int   ia = 0, ib = 0, ic = 0;
    for (int m = tid; m < N_MESH; m += 256) {
        float dx = py0 - pm[m * 2 + 0];
        float dy = py1 - pm[m * 2 + 1];
        float d = dx * dx + dy * dy;
        if (d < da)      { dc = db; ic = ib; db = da; ib = ia; da = d; ia = m; }
        else if (d < db) { dc = db; ic = ib; db = d;  ib = m; }
        else if (d < dc) { dc = d;  ic = m; }
    }
    sd[tid * 3 + 0] = da; sd[tid * 3 + 1] = db; sd[tid * 3 + 2] = dc;
    si[tid * 3 + 0] = ia; si[tid * 3 + 1] = ib; si[tid * 3 + 2] = ic;
    __syncthreads();
    if (tid == 0) {
        float b0 = 3.4e38f, b1 = 3.4e38f, b2 = 3.4e38f;
        int   j0 = 0, j1 = 0, j2 = 0;
        for (int t = 0; t < 256 * 3; ++t) {
            float d = sd[t];
            int  id = si[t];
            if (d < b0)      { b2 = b1; j2 = j1; b1 = b0; j1 = j0; b0 = d; j0 = id; }
            else if (d < b1) { b2 = b1; j2 = j1; b1 = d; j1 = id; }
            else if (d < b2) { b2 = d; j2 = id; }
        }
        float w0 = 1.f / fmaxf(b0, 1e-16f);
        float w1 = 1.f / fmaxf(b1, 1e-16f);
        float w2 = 1.f / fmaxf(b2, 1e-16f);
        float ws = w0 + w1 + w2;
        for (int j = 0; j < 3; ++j)
            zpiv[p * 3 + j] = (z[(size_t)j0 * 3 + j] * w0 + z[(size_t)j1 * 3 + j] * w1 +
                               z[(size_t)j2 * 3 + j] * w2) / ws;
    }
}

// mesh <- pivotal : one thread per mesh point, 256 pivots in LDS
__global__ __launch_bounds__(256) void knn_p2m_kernel(const float* __restrict__ zpiv,
                                                      const float* __restrict__ pp,
                                                      const float* __restrict__ pm,
                                                      float* __restrict__ xm) {
    __shared__ float sp[N_PIV * 2];
    for (int i = threadIdx.x; i < N_PIV * 2; i += 256) sp[i] = pp[i];
    __syncthreads();
    int m = blockIdx.x * 256 + threadIdx.x;
    if (m >= N_MESH) return;
    float y0 = pm[m * 2 + 0], y1 = pm[m * 2 + 1];
    float da = 3.4e38f, db = 3.4e38f, dc = 3.4e38f;
    int   ia = 0, ib = 0, ic = 0;
    for (int p = 0; p < N_PIV; ++p) {
        float dx = y0 - sp[p * 2 + 0];
        float dy = y1 - sp[p * 2 + 1];
        float d = dx * dx + dy * dy;
        if (d < da)      { dc = db; ic = ib; db = da; ib = ia; da = d; ia = p; }
        else if (d < db) { dc = db; ic = ib; db = d;  ib = p; }
        else if (d < dc) { dc = d;  ic = p; }
    }
    float w0 = 1.f / fmaxf(da, 1e-16f);
    float w1 = 1.f / fmaxf(db, 1e-16f);
    float w2 = 1.f / fmaxf(dc, 1e-16f);
    float ws = w0 + w1 + w2;
    for (int j = 0; j < 3; ++j)
        xm[(size_t)m * 3 + j] = (zpiv[ia * 3 + j] * w0 + zpiv[ib * 3 + j] * w1 +
                                 zpiv[ic * 3 + j] * w2) / ws;
}

// --------------------------------------------------------------- host side
struct MLPp { const float *W0, *W1, *W2, *b0, *b1, *b2, *g, *be; };

// norm MLP leaf order (sorted dict keys): Ws(W0,W1,W2), beta, bs(b0,b1,b2), gamma
static MLPp mlpN(void* const* din, int base) {
    MLPp p;
    p.W0 = (const float*)din[base + 0];
    p.W1 = (const float*)din[base + 1];
    p.W2 = (const float*)din[base + 2];
    p.be = (const float*)din[base + 3];
    p.b0 = (const float*)din[base + 4];
    p.b1 = (const float*)din[base + 5];
    p.b2 = (const float*)din[base + 6];
    p.g  = (const float*)din[base + 7];
    return p;
}
// plain MLP: Ws(W0,W1,W2), bs(b0,b1,b2)
static MLPp mlpP(void* const* din, int base) {
    MLPp p;
    p.W0 = (const float*)din[base + 0];
    p.W1 = (const float*)din[base + 1];
    p.W2 = (const float*)din[base + 2];
    p.b0 = (const float*)din[base + 3];
    p.b1 = (const float*)din[base + 4];
    p.b2 = (const float*)din[base + 5];
    p.g = p.be = nullptr;
    return p;
}

struct MGNp { MLPp node_enc, edge_enc, node_dec, blkE[6], blkN[6]; };

// MGN leaf order (sorted keys): blocks[0..5]{edge,node}, edge_enc, node_dec, node_enc
static MGNp mgn(void* const* din, int base) {
    MGNp g;
    for (int i = 0; i < 6; ++i) {
        g.blkE[i] = mlpN(din, base + 16 * i);
        g.blkN[i] = mlpN(din, base + 16 * i + 8);
    }
    g.edge_enc = mlpN(din, base + 96);
    g.node_dec = mlpP(din, base + 104);
    g.node_enc = mlpN(din, base + 110);
    return g;
}

static void run_mgn(const MGNp& P, const float* nfeat3, const float* efeat3, float* out3,
                    const int* src, const int* dst, const float* invdeg,
                    float* EF, float* T1, float* T2, float* NF, float* ND, float* NE,
                    hipStream_t stream) {
    const int Mn = N_MESH, Me = N_EDGES;
    dim3 B(256);
    int gN = (Mn + 63) / 64, gE = (Me + 63) / 64;
    int lnN = (Mn + 7) / 8, lnE = (Me + 7) / 8;

    // node encoder: lin3->128 relu, 128 relu, 128, LN
    gemm_k3_kernel<<<Mn / 2, B, 0, stream>>>(nfeat3, P.node_enc.W0, P.node_enc.b0, ND, Mn);
    wmma_mlp_kernel<0, 128, true><<<gN, B, 0, stream>>>(ND, nullptr, nullptr, nullptr, nullptr,
                                                        P.node_enc.W1, P.node_enc.b1, NE, Mn);
    wmma_mlp_kernel<0, 128, false><<<gN, B, 0, stream>>>(NE, nullptr, nullptr, nullptr, nullptr,
                                                         P.node_enc.W2, P.node_enc.b2, ND, Mn);
    ln128_kernel<false><<<lnN, B, 0, stream>>>(ND, P.node_enc.g, P.node_enc.be, nullptr, NF, Mn);

    // edge encoder
    gemm_k3_kernel<<<Me / 2, B, 0, stream>>>(efeat3, P.edge_enc.W0, P.edge_enc.b0, T1, Me);
    wmma_mlp_kernel<0, 128, true><<<gE, B, 0, stream>>>(T1, nullptr, nullptr, nullptr, nullptr,
                                                        P.edge_enc.W1, P.edge_enc.b1, T2, Me);
    wmma_mlp_kernel<0, 128, false><<<gE, B, 0, stream>>>(T2, nullptr, nullptr, nullptr, nullptr,
                                                         P.edge_enc.W2, P.edge_enc.b2, T1, Me);
    ln128_kernel<false><<<lnE, B, 0, stream>>>(T1, P.edge_enc.g, P.edge_enc.be, nullptr, EF, Me);

    for (int i = 0; i < 6; ++i) {
        const MLPp& Eb = P.blkE[i];
        const MLPp& Nb = P.blkN[i];
        // edge block: MLP(concat[ef, nf[src], nf[dst]]) + LN + residual
        wmma_mlp_kernel<3, 384, true><<<gE, B, 0, stream>>>(EF, NF, src, dst, nullptr,
                                                            Eb.W0, Eb.b0, T1, Me);
        wmma_mlp_kernel<0, 128, true><<<gE, B, 0, stream>>>(T1, nullptr, nullptr, nullptr, nullptr,
                                                            Eb.W1, Eb.b1, T2, Me);
        wmma_mlp_kernel<0, 128, false><<<gE, B, 0, stream>>>(T2, nullptr, nullptr, nullptr, nullptr,
                                                             Eb.W2, Eb.b2, T1, Me);
        ln128_kernel<true><<<lnE, B, 0, stream>>>(T1, Eb.g, Eb.be, EF, EF, Me);
        // mean aggregation (divide by deg fused into next gather)
        fill_zero_kernel<<<1024, B, 0, stream>>>(NE, (long)Mn * HD);
        scatter_add_kernel<<<(int)(((long)Me * HD + 255) / 256), B, 0, stream>>>(EF, dst, NE, (long)Me * HD);
        // node block: MLP(concat[nf, agg/deg]) + LN + residual
        wmma_mlp_kernel<2, 256, true><<<gN, B, 0, stream>>>(NF, NE, nullptr, nullptr, invdeg,
                                                            Nb.W0, Nb.b0, ND, Mn);
        wmma_mlp_kernel<0, 128, true><<<gN, B, 0, stream>>>(ND, nullptr, nullptr, nullptr, nullptr,
                                                            Nb.W1, Nb.b1, NE, Mn);
        wmma_mlp_kernel<0, 128, false><<<gN, B, 0, stream>>>(NE, nullptr, nullptr, nullptr, nullptr,
                                                             Nb.W2, Nb.b2, ND, Mn);
        ln128_kernel<true><<<lnN, B, 0, stream>>>(ND, Nb.g, Nb.be, NF, NF, Mn);
    }

    // node decoder: 128 relu, 128 relu, ->3
    wmma_mlp_kernel<0, 128, true><<<gN, B, 0, stream>>>(NF, nullptr, nullptr, nullptr, nullptr,
                                                        P.node_dec.W0, P.node_dec.b0, ND, Mn);
    wmma_mlp_kernel<0, 128, true><<<gN, B, 0, stream>>>(ND, nullptr, nullptr, nullptr, nullptr,
                                                        P.node_dec.W1, P.node_dec.b1, NE, Mn);
    gemm_n3_kernel<<<(Mn + 255) / 256, B, 0, stream>>>(NE, P.node_dec.W2, P.node_dec.b2, out3, Mn);
}

extern "C" void kernel_launch(void* const* d_in, const int* in_sizes, int n_in,
                              void* d_out, int out_size, void* d_ws, size_t ws_size,
                              hipStream_t stream) {
    (void)in_sizes; (void)n_in; (void)out_size; (void)ws_size;
    const float* x        = (const float*)d_in[0];
    const float* efeat    = (const float*)d_in[1];
    const float* pos_mesh = (const float*)d_in[2];
    const float* pos_piv  = (const float*)d_in[3];
    MGNp enc = mgn(d_in, 4);
    MGNp dec = mgn(d_in, 122);
    const float* piv_g = (const float*)d_in[240];
    const float* piv_b = (const float*)d_in[241];
    const int* eidx = (const int*)d_in[242];
    const int* src = eidx;
    const int* dst = eidx + N_EDGES;

    float* f = (float*)d_ws;
    float* EF  = f; f += (size_t)N_EDGES * HD;
    float* T1  = f; f += (size_t)N_EDGES * HD;
    float* T2  = f; f += (size_t)N_EDGES * HD;
    float* NF  = f; f += (size_t)N_MESH * HD;
    float* ND  = f; f += (size_t)N_MESH * HD;
    float* NE  = f; f += (size_t)N_MESH * HD;
    float* DEG = f; f += N_MESH;
    float* Z   = f; f += (size_t)N_MESH * 3;
    float* ZP  = f; f += N_PIV * 3;
    float* XM  = f; f += (size_t)N_MESH * 3;

    dim3 B(256);
    // degrees (shared by both MGNs)
    fill_zero_kernel<<<256, B, 0, stream>>>(DEG, N_MESH);
    deg_kernel<<<(N_EDGES + 255) / 256, B, 0, stream>>>(dst, DEG, N_EDGES);
    invdeg_kernel<<<(N_MESH + 255) / 256, B, 0, stream>>>(DEG, N_MESH);

    // encode
    run_mgn(enc, x, efeat, Z, src, dst, DEG, EF, T1, T2, NF, ND, NE, stream);
    ln3_kernel<<<(N_MESH + 255) / 256, B, 0, stream>>>(Z, piv_g, piv_b, N_MESH);
    knn_m2p_kernel<<<N_PIV, B, 0, stream>>>(Z, pos_mesh, pos_piv, ZP);
    knn_p2m_kernel<<<(N_MESH + 255) / 256, B, 0, stream>>>(ZP, pos_piv, pos_mesh, XM);

    // decode
    run_mgn(dec, XM, efeat, (float*)d_out, src, dst, DEG, EF, T1, T2, NF, ND, NE, stream);
}